// HeteroGNN_42537356100337
// MI455X (gfx1250) — compile-verified
//
#include <hip/hip_runtime.h>
#include <hip/hip_bf16.h>

// Problem constants (match reference)
#define N_L   100000
#define D_IN  64
#define HCH   64
#define D_OUT 32
#define E_LL  1000000

typedef float v2f __attribute__((ext_vector_type(2)));
typedef float v8f __attribute__((ext_vector_type(8)));

// ---------------------------------------------------------------- utilities
__global__ void zero_kernel(float* __restrict__ p, int n) {
    int i = blockIdx.x * blockDim.x + threadIdx.x;
    if (i < n) p[i] = 0.0f;
}

// deg[c] += 1 for every edge destination
__global__ void deg_kernel(const int* __restrict__ col, float* __restrict__ deg, int e) {
    int i = blockIdx.x * blockDim.x + threadIdx.x;
    if (i < e) atomicAdd(&deg[col[i]], 1.0f);
}

// in-place deg -> dinv = deg>0 ? rsqrt(deg) : 0
__global__ void dinv_kernel(float* __restrict__ deg, int n) {
    int i = blockIdx.x * blockDim.x + threadIdx.x;
    if (i < n) {
        float d = deg[i];
        deg[i] = (d > 0.0f) ? rsqrtf(d) : 0.0f;
    }
}

// ------------------------------------------------- GEMM1: h = x @ W_gcn
// x: [N_L,64] row-major, W: [64,64] row-major, h: [N_L,64]
// One wave (32 lanes) per 16-row M tile; 4 N-tiles of 16; K stepped by 4
// using V_WMMA_F32_16X16X4_F32 (exact fp32).
__global__ __launch_bounds__(256)
void gemm_xw_kernel(const float* __restrict__ x, const float* __restrict__ W,
                    float* __restrict__ h) {
    __shared__ float Ws[64 * 64];
    for (int i = threadIdx.x; i < 64 * 64; i += 256) Ws[i] = W[i];
    __syncthreads();

    const int wave = threadIdx.x >> 5;   // wave32
    const int lane = threadIdx.x & 31;
    const int tile = blockIdx.x * 8 + wave;
    if (tile >= N_L / 16) return;        // wave-uniform guard -> EXEC all 1s

    const int m0 = tile * 16;
    const int lm = lane & 15;            // M (A) / N (B,C) index within tile
    const int hi = lane >> 4;            // lane-half selects K pair / M half

    const float* xrow = x + (size_t)(m0 + lm) * 64;
    v8f acc[4] = {};

    for (int k0 = 0; k0 < 64; k0 += 4) {
        const int ka = k0 + hi * 2;      // this lane-half's K base
        v2f a;
        a.x = xrow[ka];
        a.y = xrow[ka + 1];
#pragma unroll
        for (int nt = 0; nt < 4; ++nt) {
            v2f b;
            b.x = Ws[ka * 64 + nt * 16 + lm];
            b.y = Ws[(ka + 1) * 64 + nt * 16 + lm];
            acc[nt] = __builtin_amdgcn_wmma_f32_16x16x4_f32(
                false, a, false, b, (short)0, acc[nt], false, false);
        }
    }

#pragma unroll
    for (int nt = 0; nt < 4; ++nt)
#pragma unroll
        for (int j = 0; j < 8; ++j)
            h[(size_t)(m0 + j + hi * 8) * 64 + nt * 16 + lm] = acc[nt][j];
}

// ------------------------------------- scatter: agg[c] += h[r] * dinv[r]*dinv[c]
// 64 threads per edge (one per feature); edge id is wave-uniform so the
// index/dinv loads broadcast and the h gather / atomic adds are coalesced.
__global__ __launch_bounds__(256)
void scatter_kernel(const int* __restrict__ row, const int* __restrict__ col,
                    const float* __restrict__ h, const float* __restrict__ dinv,
                    float* __restrict__ agg) {
    int t = blockIdx.x * blockDim.x + threadIdx.x;
    int e = t >> 6;
    int j = t & 63;
    if (e >= E_LL) return;
    int r = row[e];
    int c = col[e];
    float nrm = dinv[r] * dinv[c];
    atomicAdd(&agg[(size_t)c * 64 + j], h[(size_t)r * 64 + j] * nrm);
}

// -------------------------- GEMM2: out = relu(agg + b_gcn) @ W_lin + b_lin
// agg: [N_L,64], W_lin: [64,32], out: [N_L,32]. relu+bias fused into A load.
__global__ __launch_bounds__(256)
void gemm_out_kernel(const float* __restrict__ agg, const float* __restrict__ bgcn,
                     const float* __restrict__ Wlin, const float* __restrict__ blin,
                     float* __restrict__ out) {
    __shared__ float Ws[64 * 32];
    __shared__ float bg[64];
    __shared__ float bl[32];
    for (int i = threadIdx.x; i < 64 * 32; i += 256) Ws[i] = Wlin[i];
    if (threadIdx.x < 64) bg[threadIdx.x] = bgcn[threadIdx.x];
    if (threadIdx.x < 32) bl[threadIdx.x] = blin[threadIdx.x];
    __syncthreads();

    const int wave = threadIdx.x >> 5;
    const int lane = threadIdx.x & 31;
    const int tile = blockIdx.x * 8 + wave;
    if (tile >= N_L / 16) return;

    const int m0 = tile * 16;
    const int lm = lane & 15;
    const int hi = lane >> 4;

    const float* arow = agg + (size_t)(m0 + lm) * 64;
    v8f acc[2] = {};

    for (int k0 = 0; k0 < 64; k0 += 4) {
        const int ka = k0 + hi * 2;
        float a0 = arow[ka]     + bg[ka];
        float a1 = arow[ka + 1] + bg[ka + 1];
        v2f a;
        a.x = a0 > 0.0f ? a0 : 0.0f;   // fused ReLU
        a.y = a1 > 0.0f ? a1 : 0.0f;
#pragma unroll
        for (int nt = 0; nt < 2; ++nt) {
            v2f b;
            b.x = Ws[ka * 32 + nt * 16 + lm];
            b.y = Ws[(ka + 1) * 32 + nt * 16 + lm];
            acc[nt] = __builtin_amdgcn_wmma_f32_16x16x4_f32(
                false, a, false, b, (short)0, acc[nt], false, false);
        }
    }

#pragma unroll
    for (int nt = 0; nt < 2; ++nt)
#pragma unroll
        for (int j = 0; j < 8; ++j)
            out[(size_t)(m0 + j + hi * 8) * 32 + nt * 16 + lm] =
                acc[nt][j] + bl[nt * 16 + lm];
}

// ---------------------------------------------------------------- launcher
extern "C" void kernel_launch(void* const* d_in, const int* in_sizes, int n_in,
                              void* d_out, int out_size, void* d_ws, size_t ws_size,
                              hipStream_t stream) {
    (void)in_sizes; (void)n_in; (void)out_size; (void)ws_size;

    const float* x_local = (const float*)d_in[0];
    const float* W_gcn   = (const float*)d_in[2];
    const float* b_gcn   = (const float*)d_in[3];
    const float* W_lin   = (const float*)d_in[10];
    const float* b_lin   = (const float*)d_in[11];
    const int*   edge_ll = (const int*)d_in[12];
    const int*   row     = edge_ll;          // edge_ll[0]
    const int*   col     = edge_ll + E_LL;   // edge_ll[1]

    // Workspace layout (floats): dinv[N_L] | h[N_L*64] | agg[N_L*64] ~= 51.6 MB
    float* ws   = (float*)d_ws;
    float* dinv = ws;
    float* h    = ws + N_L;
    float* agg  = ws + N_L + (size_t)N_L * 64;
    float* out  = (float*)d_out;

    const int gemmBlocks = (N_L / 16 + 7) / 8;   // 8 waves (M tiles) per block

    zero_kernel<<<(N_L + 255) / 256, 256, 0, stream>>>(dinv, N_L);
    zero_kernel<<<(N_L * 64 + 255) / 256, 256, 0, stream>>>(agg, N_L * 64);
    deg_kernel<<<(E_LL + 255) / 256, 256, 0, stream>>>(col, dinv, E_LL);
    dinv_kernel<<<(N_L + 255) / 256, 256, 0, stream>>>(dinv, N_L);
    gemm_xw_kernel<<<gemmBlocks, 256, 0, stream>>>(x_local, W_gcn, h);
    scatter_kernel<<<(E_LL * 64) / 256, 256, 0, stream>>>(row, col, h, dinv, agg);
    gemm_out_kernel<<<gemmBlocks, 256, 0, stream>>>(agg, b_gcn, W_lin, b_lin, out);
}